// ProbAttention_20572893347882
// MI455X (gfx1250) — compile-verified
//
#include <hip/hip_runtime.h>
#include <hip/hip_bf16.h>
#include <float.h>

typedef __attribute__((ext_vector_type(2))) float v2f;
typedef __attribute__((ext_vector_type(8))) float v8f;

#define B_ 4
#define L_ 2048
#define H_ 8
#define D_ 64
#define SK_ 40
#define U_ 40
#define UPAD_ 48
#define NCHUNK_ 32
#define CH_ (L_ / NCHUNK_) /* 64 */
#define NT_ (L_ / 16)      /* 128 */
#define SPLIT_ 4           /* key-range splits inside one attention block */
#define JT_ (NT_ / SPLIT_) /* 32 tiles per split */

// ---------------------------------------------------------------------------
// K1: M[b,h,i] = max_s(q_i . k_{idx[i,s]}) - sum_s(...) / L
// One wave32 per query; each lane owns 2 of the 64 dims.
// ---------------------------------------------------------------------------
__global__ __launch_bounds__(256) void k_metric(const float* __restrict__ q,
                                                const float* __restrict__ k,
                                                const long long* __restrict__ idxs,
                                                float* __restrict__ Mbuf) {
  int wave = blockIdx.x * (blockDim.x >> 5) + (threadIdx.x >> 5);
  int lane = threadIdx.x & 31;
  int i = wave % L_;
  int bh = wave / L_;
  int h = bh % H_;
  int b = bh / H_;
  const float* qp = q + ((size_t)(b * L_ + i) * H_ + h) * D_ + 2 * lane;
  float qx = qp[0], qy = qp[1];
  float mx = -FLT_MAX, sm = 0.f;
  for (int s = 0; s < SK_; ++s) {
    int kidx = (int)idxs[(size_t)i * SK_ + s];
    const float* kp = k + ((size_t)(b * L_ + kidx) * H_ + h) * D_ + 2 * lane;
    float d = qx * kp[0] + qy * kp[1];
    #pragma unroll
    for (int off = 16; off; off >>= 1) d += __shfl_xor(d, off, 32);
    mx = fmaxf(mx, d);
    sm += d;
  }
  if (lane == 0) Mbuf[(size_t)bh * L_ + i] = mx - sm * (1.0f / (float)L_);
}

// ---------------------------------------------------------------------------
// K2: top-40 indices of M per (b,h) via iterative argmax (lower-index ties).
// ---------------------------------------------------------------------------
__global__ __launch_bounds__(256) void k_topk(const float* __restrict__ Mbuf,
                                              int* __restrict__ Mtop) {
  __shared__ float vals[L_];
  __shared__ float rv[256];
  __shared__ int ri[256];
  int bh = blockIdx.x;
  int tid = threadIdx.x;
  for (int t = tid; t < L_; t += 256) vals[t] = Mbuf[(size_t)bh * L_ + t];
  __syncthreads();
  for (int t = 0; t < U_; ++t) {
    float bv = -FLT_MAX;
    int bi = 0x7fffffff;
    for (int c = tid; c < L_; c += 256) {
      float v = vals[c];
      if (v > bv || (v == bv && c < bi)) { bv = v; bi = c; }
    }
    rv[tid] = bv; ri[tid] = bi;
    __syncthreads();
    for (int s = 128; s > 0; s >>= 1) {
      if (tid < s) {
        float ov = rv[tid + s]; int oi = ri[tid + s];
        if (ov > rv[tid] || (ov == rv[tid] && oi < ri[tid])) { rv[tid] = ov; ri[tid] = oi; }
      }
      __syncthreads();
    }
    if (tid == 0) { Mtop[bh * U_ + t] = ri[0]; vals[ri[0]] = -FLT_MAX; }
    __syncthreads();
  }
}

// ---------------------------------------------------------------------------
// K3a/b/c: cumsum of V over L (per b,h,d) -> default output (same layout).
// ---------------------------------------------------------------------------
__global__ __launch_bounds__(64) void k_cumsum_part(const float* __restrict__ vv,
                                                    float* __restrict__ part) {
  int blk = blockIdx.x;
  int chunk = blk & (NCHUNK_ - 1);
  int bh = blk / NCHUNK_;
  int h = bh % H_, b = bh / H_;
  int d = threadIdx.x;
  size_t base = ((size_t)(b * L_ + chunk * CH_) * H_ + h) * D_ + d;
  float s = 0.f;
  for (int l = 0; l < CH_; ++l) s += vv[base + (size_t)l * H_ * D_];
  part[((size_t)bh * NCHUNK_ + chunk) * D_ + d] = s;
}

__global__ __launch_bounds__(64) void k_cumsum_scan(float* __restrict__ part) {
  int bh = blockIdx.x;
  int d = threadIdx.x;
  float run = 0.f;
  for (int c = 0; c < NCHUNK_; ++c) {
    size_t idx = ((size_t)bh * NCHUNK_ + c) * D_ + d;
    float t = part[idx];
    part[idx] = run;
    run += t;
  }
}

__global__ __launch_bounds__(64) void k_cumsum_write(const float* __restrict__ vv,
                                                     const float* __restrict__ part,
                                                     float* __restrict__ outp) {
  int blk = blockIdx.x;
  int chunk = blk & (NCHUNK_ - 1);
  int bh = blk / NCHUNK_;
  int h = bh % H_, b = bh / H_;
  int d = threadIdx.x;
  size_t base = ((size_t)(b * L_ + chunk * CH_) * H_ + h) * D_ + d;
  float acc = part[((size_t)bh * NCHUNK_ + chunk) * D_ + d];
  for (int l = 0; l < CH_; ++l) {
    acc += vv[base + (size_t)l * H_ * D_];
    outp[base + (size_t)l * H_ * D_] = acc;
  }
}

// ---------------------------------------------------------------------------
// K4: flash-style masked attention on the 40 selected queries using
// V_WMMA_F32_16X16X4_F32. One block per (b,h); 12 waves = 4 key-range
// split-groups x 3 row-tiles. Each split keeps its own online-softmax
// state; exact (m,l,O) merge at the end through LDS.
//
// LDS is time-shared: loop phase uses K/V tiles + P scratch (11584 floats),
// combine phase reuses the same buffer for partial O/m/l (12672 floats).
// ---------------------------------------------------------------------------
#define SMEM_F 12672
#define SK_OFF 0      /* sK: 4 x 1040 floats                  */
#define SV_OFF 4160   /* sV: 4 x 1040 floats                  */
#define SP_OFF 8320   /* sP: 12 x 272 floats                  */
#define SO_OFF 0      /* combine: 12 x 1024 floats            */
#define SM_OFF 12288  /* combine: 12 x 16 row maxes           */
#define SL_OFF 12480  /* combine: 12 x 16 row sums            */

__global__ __launch_bounds__(384) void k_attn(const float* __restrict__ q,
                                              const float* __restrict__ kk,
                                              const float* __restrict__ vv,
                                              const int* __restrict__ Mtop,
                                              float* __restrict__ outp) {
  __shared__ float smem[SMEM_F];
  __shared__ int sTop[UPAD_];

  const int bh = blockIdx.x;
  const int h = bh % H_, b = bh / H_;
  const int tid = threadIdx.x;
  const int wv = tid >> 5;
  const int lane = tid & 31;
  const int sg = wv / 3;        // split group 0..3 (keys [sg*512, sg*512+512))
  const int wt = wv % 3;        // row tile 0..2
  const int tsg = tid - sg * 96;
  const int hi = lane >> 4;
  const int n16 = lane & 15;
  const int koff = hi ? 2 : 0;

  float* sK = smem + SK_OFF + sg * 1040;
  float* sV = smem + SV_OFF + sg * 1040;
  float* sP = smem + SP_OFF + wv * 272;

  if (tid < UPAD_) sTop[tid] = (tid < U_) ? Mtop[bh * U_ + tid] : 0x7fffffff;
  __syncthreads();

  // A fragments: Q_reduce rows, scale 1/sqrt(D)=0.125 folded in.
  v2f a[16];
  {
    int urow = wt * 16 + n16;  // M index of this lane (both halves cover M=0..15)
    if (urow < U_) {
      const float* qp = q + ((size_t)(b * L_ + sTop[urow]) * H_ + h) * D_;
      #pragma unroll
      for (int c = 0; c < 16; ++c) {
        a[c].x = 0.125f * qp[4 * c + koff];
        a[c].y = 0.125f * qp[4 * c + koff + 1];
      }
    } else {
      #pragma unroll
      for (int c = 0; c < 16; ++c) { a[c].x = 0.f; a[c].y = 0.f; }
    }
  }

  int row_top[8];
  #pragma unroll
  for (int r = 0; r < 8; ++r) {
    int u = wt * 16 + r + 8 * hi;            // C/D layout: M = r + 8*half
    row_top[r] = (u < U_) ? sTop[u] : 0x7fffffff;
  }

  float m_run[8], l_run[8];
  #pragma unroll
  for (int r = 0; r < 8; ++r) { m_run[r] = -1e30f; l_run[r] = 0.f; }
  v8f oacc[4];
  #pragma unroll
  for (int t = 0; t < 4; ++t) oacc[t] = (v8f){};

  const int jbeg = sg * JT_;
  const int jend = jbeg + JT_;
  for (int j = jbeg; j < jend; ++j) {
    // Stage this split's K and V 16x64 tiles (coalesced, 96 threads per group).
    for (int t = tsg; t < 16 * 64; t += 96) {
      int rr = t >> 6, e = t & 63;
      size_t g = ((size_t)(b * L_ + j * 16 + rr) * H_ + h) * D_ + e;
      sK[rr * 65 + e] = kk[g];
      sV[rr * 65 + e] = vv[g];
    }
    // Prefetch next tile's rows (global_prefetch_b8) to hide fetch latency.
    if (j + 1 < jend && tsg < 16) {
      size_t g = ((size_t)(b * L_ + (j + 1) * 16 + tsg) * H_ + h) * D_;
      __builtin_prefetch(&kk[g], 0, 0);
      __builtin_prefetch(&kk[g + 32], 0, 0);
      __builtin_prefetch(&vv[g], 0, 0);
      __builtin_prefetch(&vv[g + 32], 0, 0);
    }
    __syncthreads();

    // Preload all 16 B fragments, then run the WMMA chain back-to-back.
    v2f bb[16];
    #pragma unroll
    for (int c = 0; c < 16; ++c) {
      bb[c].x = sK[n16 * 65 + 4 * c + koff];
      bb[c].y = sK[n16 * 65 + 4 * c + koff + 1];
    }
    v8f sc = (v8f){};
    #pragma unroll
    for (int c = 0; c < 16; ++c)
      sc = __builtin_amdgcn_wmma_f32_16x16x4_f32(false, a[c], false, bb[c],
                                                 (short)0, sc, false, false);

    // Causal mask + online softmax. Butterfly steps batched across all 8
    // rows so the 8 independent cross-lane ops pipeline under one wait.
    int col = j * 16 + n16;
    float sval[8], red[8];
    #pragma unroll
    for (int r = 0; r < 8; ++r) {
      float v = sc[r];
      v = (col > row_top[r]) ? -1e30f : v;
      sval[r] = v;
      red[r] = v;
    }
    #pragma unroll
    for (int off = 1; off < 16; off <<= 1) {
      float tr[8];
      #pragma unroll
      for (int r = 0; r < 8; ++r) tr[r] = __shfl_xor(red[r], off, 32);
      #pragma unroll
      for (int r = 0; r < 8; ++r) red[r] = fmaxf(red[r], tr[r]);
    }
    float p[8], alpha[8], rs[8];
    #pragma unroll
    for (int r = 0; r < 8; ++r) {
      float nm = fmaxf(m_run[r], red[r]);
      alpha[r] = __expf(m_run[r] - nm);
      p[r] = __expf(sval[r] - nm);
      rs[r] = p[r];
      m_run[r] = nm;
    }
    #pragma unroll
    for (int off = 1; off < 16; off <<= 1) {
      float tr[8];
      #pragma unroll
      for (int r = 0; r < 8; ++r) tr[r] = __shfl_xor(rs[r], off, 32);
      #pragma unroll
      for (int r = 0; r < 8; ++r) rs[r] += tr[r];
    }
    #pragma unroll
    for (int r = 0; r < 8; ++r) l_run[r] = l_run[r] * alpha[r] + rs[r];
    #pragma unroll
    for (int t = 0; t < 4; ++t)
      #pragma unroll
      for (int r = 0; r < 8; ++r) oacc[t][r] *= alpha[r];

    // Re-layout attn tile C->A via wave-private LDS.
    #pragma unroll
    for (int r = 0; r < 8; ++r) sP[(r + 8 * hi) * 17 + n16] = p[r];
    __syncthreads();

    v2f ap[4];
    #pragma unroll
    for (int c = 0; c < 4; ++c) {
      ap[c].x = sP[n16 * 17 + 4 * c + koff];
      ap[c].y = sP[n16 * 17 + 4 * c + koff + 1];
    }
    // Preload all 16 PV B fragments, then 16 WMMAs back-to-back.
    v2f bv[16];
    #pragma unroll
    for (int t = 0; t < 4; ++t)
      #pragma unroll
      for (int c = 0; c < 4; ++c) {
        bv[t * 4 + c].x = sV[(4 * c + koff) * 65 + t * 16 + n16];
        bv[t * 4 + c].y = sV[(4 * c + koff + 1) * 65 + t * 16 + n16];
      }
    #pragma unroll
    for (int t = 0; t < 4; ++t)
      #pragma unroll
      for (int c = 0; c < 4; ++c)
        oacc[t] = __builtin_amdgcn_wmma_f32_16x16x4_f32(false, ap[c], false,
                                                        bv[t * 4 + c], (short)0,
                                                        oacc[t], false, false);
    __syncthreads();  // protect LDS tiles before next iteration's loads
  }

  // ---- exact flash merge of the 4 split-group partial states ----
  // (Last loop barrier guarantees all K/V/P reads are complete; the combine
  //  buffers below alias that space.)
  float* sO = smem + SO_OFF;
  float* sM = smem + SM_OFF;
  float* sL = smem + SL_OFF;
  const int tile = sg * 3 + wt;
  #pragma unroll
  for (int t = 0; t < 4; ++t)
    #pragma unroll
    for (int r = 0; r < 8; ++r)
      sO[(tile * 16 + r + 8 * hi) * 64 + t * 16 + n16] = oacc[t][r];
  if (n16 == 0) {
    #pragma unroll
    for (int r = 0; r < 8; ++r) {
      sM[tile * 16 + r + 8 * hi] = m_run[r];
      sL[tile * 16 + r + 8 * hi] = l_run[r];
    }
  }
  __syncthreads();

  if (sg == 0) {
    #pragma unroll
    for (int r = 0; r < 8; ++r) {
      int M = r + 8 * hi;
      int u = wt * 16 + M;
      float m0 = sM[(0 * 3 + wt) * 16 + M];
      float m1 = sM[(1 * 3 + wt) * 16 + M];
      float m2 = sM[(2 * 3 + wt) * 16 + M];
      float m3 = sM[(3 * 3 + wt) * 16 + M];
      float mt = fmaxf(fmaxf(m0, m1), fmaxf(m2, m3));
      float f0 = __expf(m0 - mt), f1 = __expf(m1 - mt);
      float f2 = __expf(m2 - mt), f3 = __expf(m3 - mt);
      float lt = sL[(0 * 3 + wt) * 16 + M] * f0 + sL[(1 * 3 + wt) * 16 + M] * f1 +
                 sL[(2 * 3 + wt) * 16 + M] * f2 + sL[(3 * 3 + wt) * 16 + M] * f3;
      if (u < U_) {
        int row = sTop[u];
        #pragma unroll
        for (int t = 0; t < 4; ++t) {
          float o = sO[((0 * 3 + wt) * 16 + M) * 64 + t * 16 + n16] * f0 +
                    sO[((1 * 3 + wt) * 16 + M) * 64 + t * 16 + n16] * f1 +
                    sO[((2 * 3 + wt) * 16 + M) * 64 + t * 16 + n16] * f2 +
                    sO[((3 * 3 + wt) * 16 + M) * 64 + t * 16 + n16] * f3;
          outp[((size_t)(b * L_ + row) * H_ + h) * D_ + t * 16 + n16] = o / lt;
        }
      }
    }
  }
}

// ---------------------------------------------------------------------------
extern "C" void kernel_launch(void* const* d_in, const int* in_sizes, int n_in,
                              void* d_out, int out_size, void* d_ws, size_t ws_size,
                              hipStream_t stream) {
  const float* q = (const float*)d_in[0];
  const float* k = (const float*)d_in[1];
  const float* v = (const float*)d_in[2];
  const long long* idxs = (const long long*)d_in[3];  // int64 indices
  float* outp = (float*)d_out;

  // Workspace layout (all freshly written each call):
  float* Mbuf = (float*)d_ws;                          // B*H*L floats
  int* Mtop = (int*)(Mbuf + (size_t)B_ * H_ * L_);     // B*H*U ints
  float* part = (float*)(Mtop + B_ * H_ * U_);         // B*H*NCHUNK*D floats

  k_metric<<<(B_ * H_ * L_) / 8, 256, 0, stream>>>(q, k, idxs, Mbuf);
  k_topk<<<B_ * H_, 256, 0, stream>>>(Mbuf, Mtop);
  k_cumsum_part<<<B_ * H_ * NCHUNK_, 64, 0, stream>>>(v, part);
  k_cumsum_scan<<<B_ * H_, 64, 0, stream>>>(part);
  k_cumsum_write<<<B_ * H_ * NCHUNK_, 64, 0, stream>>>(v, part, outp);
  k_attn<<<B_ * H_, 384, 0, stream>>>(q, k, v, Mtop, outp);
}